// ann1_snn2_6176162971991
// MI455X (gfx1250) — compile-verified
//
#include <hip/hip_runtime.h>

// CDNA5 wave32 WMMA fragment vector types
typedef float v2f __attribute__((ext_vector_type(2)));
typedef float v8f __attribute__((ext_vector_type(8)));

#define T_LEN   100
#define BATCH   512
#define NN      500
#define NPAD    512
#define NC      10
#define ALPHA_M 0.778800783071404878f   // exp(-1/4)
#define THRESH  1.0f

// ---------------------------------------------------------------------------
// Kernel 1: rate coding  coding = sigmoid(inputs)
// ---------------------------------------------------------------------------
__global__ void sigmoid_kernel(const float* __restrict__ x,
                               float* __restrict__ y, int n) {
  int i = blockIdx.x * blockDim.x + threadIdx.x;
  if (i < n) y[i] = 1.0f / (1.0f + __expf(-x[i]));
}

// ---------------------------------------------------------------------------
// Kernel 2: time-hoisted GEMM  G[b,o] = sum_i coding[b,i] * W2[o,i]
// One wave per 16x16 tile; exact fp32 tensor path V_WMMA_F32_16X16X4_F32.
// Branch-free inner loop: B row pointer clamped, zero-mask applied at store
// (D column == B column == lane&15, so a per-lane store mask is exact).
// Two independent accumulator chains hide the WMMA D->C RAW latency.
// G stored padded [512][512]; pad columns come out exactly 0.
// ---------------------------------------------------------------------------
__global__ void gemm1_wmma_kernel(const float* __restrict__ coding,
                                  const float* __restrict__ W2,
                                  float* __restrict__ G) {
  const int mt  = blockIdx.x;          // batch tile
  const int nt  = blockIdx.y;          // output-neuron tile
  const int l   = threadIdx.x;         // lane 0..31
  const int row = l & 15;
  const int kh  = l >> 4;              // K-half of the 16x4 fragment
  const int am  = mt * 16 + row;       // A row (batch)
  const int bn  = nt * 16 + row;       // B col (output neuron)
  const int bc  = (bn < NN) ? bn : (NN - 1);      // clamped, always valid
  const float bmask = (bn < NN) ? 1.0f : 0.0f;    // applied at store

  const float* __restrict__ ap = coding + (size_t)am * NN;
  const float* __restrict__ bp = W2     + (size_t)bc * NN;

  v8f acc0 = {}, acc1 = {};
  for (int k = 0; k < 496; k += 8) {   // 62 double-chunks
    const int ka = k + 2 * kh;
    v2f a0, b0, a1, b1;
    a0[0] = ap[ka];     a0[1] = ap[ka + 1];
    b0[0] = bp[ka];     b0[1] = bp[ka + 1];
    a1[0] = ap[ka + 4]; a1[1] = ap[ka + 5];
    b1[0] = bp[ka + 4]; b1[1] = bp[ka + 5];
    acc0 = __builtin_amdgcn_wmma_f32_16x16x4_f32(
        false, a0, false, b0, (short)0, acc0, false, false);
    acc1 = __builtin_amdgcn_wmma_f32_16x16x4_f32(
        false, a1, false, b1, (short)0, acc1, false, false);
  }
  {                                    // tail chunk k = 496..499
    const int ka = 496 + 2 * kh;
    v2f a0, b0;
    a0[0] = ap[ka]; a0[1] = ap[ka + 1];
    b0[0] = bp[ka]; b0[1] = bp[ka + 1];
    acc0 = __builtin_amdgcn_wmma_f32_16x16x4_f32(
        false, a0, false, b0, (short)0, acc0, false, false);
  }

  // D layout: VGPR r -> M = r + 8*kh, N = row
  const int gm = mt * 16 + 8 * kh;
  const int gn = nt * 16 + row;
#pragma unroll
  for (int r = 0; r < 8; ++r)
    G[(gm + r) * NPAD + gn] = (acc0[r] + acc1[r]) * bmask;
}

// ---------------------------------------------------------------------------
// Kernel 3: persistent SNN time loop.
// Grid: 32 workgroups (16 batch rows each) x 512 threads (16 waves).
// Wave w owns K-slice [32w, 32w+32). All recurrent state lives in VGPRs in
// WMMA-A layout; per step each wave issues 8x v_wmma_f32_16x16x4_f32 on two
// independent accumulator chains against preloaded W3 B-fragments.
// Parity-double-buffered LDS partials -> one barrier per timestep.
// ---------------------------------------------------------------------------
__global__ void snn_persistent_kernel(const float* __restrict__ G,
                                      const float* __restrict__ b2,
                                      const float* __restrict__ a1_3,
                                      const float* __restrict__ a2_3,
                                      const float* __restrict__ W3,
                                      const float* __restrict__ b3,
                                      const float* __restrict__ a1_2,
                                      const float* __restrict__ a2_2,
                                      float* __restrict__ out) {
  __shared__ float partial[2][16 * 256];   // 2 x (16 waves x 8 VGPRs x 32 lanes)

  const int tid   = threadIdx.x;
  const int w     = tid >> 5;           // wave id 0..15
  const int l     = tid & 31;           // lane
  const int row   = l & 15;             // m (batch row) for A / class for B,D
  const int kh    = l >> 4;
  const int bbase = blockIdx.x * 16;
  const int brow  = bbase + row;

  // Per-lane state, 16 elements each, in A-fragment layout:
  // element e = 2*c + j  <->  k = 32*w + 4*c + 2*kh + j
  float g[16], b2v[16], a1v[16], a2v[16], vcar[16], y1[16], y2[16];
  v2f   w3f[8];
#pragma unroll
  for (int c = 0; c < 8; ++c) {
#pragma unroll
    for (int j = 0; j < 2; ++j) {
      const int n  = w * 32 + c * 4 + 2 * kh + j;   // < 512
      const int e  = c * 2 + j;
      const bool ok = (n < NN);
      g[e]    = G[brow * NPAD + n];                 // pad cols are 0
      b2v[e]  = ok ? b2[n]   : 0.0f;
      a1v[e]  = ok ? a1_3[n] : 0.0f;
      a2v[e]  = ok ? a2_3[n] : 0.0f;
      vcar[e] = 0.0f; y1[e] = 0.0f; y2[e] = 0.0f;
      // B fragment for GEMM2: B(k=n, col=row) = W3[row, n]
      w3f[c][j] = (ok && row < NC) ? W3[row * NN + n] : 0.0f;
    }
  }

  // scalar recurrence for k[t] (axon2 coefficients are uniform by construction)
  const float a1k = a1_2[0];
  const float a2k = a2_2[0];
  float kp1 = 0.0f, kp2 = 0.0f;

  // wave-0 LIF state for layer 3 (D layout: m = r + 8*kh, class = row)
  float v3car[8];
#pragma unroll
  for (int r = 0; r < 8; ++r) v3car[r] = 0.0f;
  const float b3v = (row < NC) ? b3[row] : 0.0f;

  for (int t = 0; t < T_LEN; ++t) {
    const float kt = a1k * kp1 + a2k * kp2 + 1.0f;  // k[0]=1
    kp2 = kp1; kp1 = kt;

    v8f acc0 = {}, acc1 = {};
#pragma unroll
    for (int c = 0; c < 8; ++c) {
      v2f afrag;
#pragma unroll
      for (int j = 0; j < 2; ++j) {
        const int e = c * 2 + j;
        // LIF layer 2: psp2 = k[t]*G + b2 ; v = alpha*v_carry + psp2
        float v = ALPHA_M * vcar[e] + kt * g[e] + b2v[e];
        float s = (v > THRESH) ? 1.0f : 0.0f;
        vcar[e] = v * (1.0f - s);                   // reset-to-zero carry
        // axon3 dual-exp IIR on spikes
        float yn = a1v[e] * y1[e] + a2v[e] * y2[e] + s;
        y2[e] = y1[e]; y1[e] = yn;
        afrag[j] = yn;
      }
      if (c & 1)
        acc1 = __builtin_amdgcn_wmma_f32_16x16x4_f32(
            false, afrag, false, w3f[c], (short)0, acc1, false, false);
      else
        acc0 = __builtin_amdgcn_wmma_f32_16x16x4_f32(
            false, afrag, false, w3f[c], (short)0, acc0, false, false);
    }

    // cross-wave K reduction through parity-double-buffered LDS
    const int buf = t & 1;
#pragma unroll
    for (int r = 0; r < 8; ++r)
      partial[buf][w * 256 + r * 32 + l] = acc0[r] + acc1[r];
    __syncthreads();   // single barrier per step; see double-buffer argument

    if (w == 0) {
#pragma unroll
      for (int r = 0; r < 8; ++r) {
        float p = 0.0f;
#pragma unroll
        for (int ww = 0; ww < 16; ++ww) p += partial[buf][ww * 256 + r * 32 + l];
        // LIF layer 3
        float v3 = ALPHA_M * v3car[r] + p + b3v;
        float s3 = (v3 > THRESH) ? 1.0f : 0.0f;
        v3car[r] = v3 * (1.0f - s3);
        if (row < NC) {
          const int m = r + 8 * kh;                 // batch row in tile
          out[((bbase + m) * NC + row) * T_LEN + t] = s3;
        }
      }
    }
  }
}

// ---------------------------------------------------------------------------
extern "C" void kernel_launch(void* const* d_in, const int* in_sizes, int n_in,
                              void* d_out, int out_size, void* d_ws, size_t ws_size,
                              hipStream_t stream) {
  const float* inputs = (const float*)d_in[0];
  const float* a1_2   = (const float*)d_in[1];
  const float* a2_2   = (const float*)d_in[2];
  const float* W2     = (const float*)d_in[3];
  const float* b2     = (const float*)d_in[4];
  const float* a1_3   = (const float*)d_in[5];
  const float* a2_3   = (const float*)d_in[6];
  const float* W3     = (const float*)d_in[7];
  const float* b3     = (const float*)d_in[8];
  float* out = (float*)d_out;

  float* coding = (float*)d_ws;                                               // 512*500 f32
  float* G      = (float*)((char*)d_ws + (size_t)BATCH * NN * sizeof(float)); // 512*512 f32

  const int n = BATCH * NN;
  sigmoid_kernel<<<(n + 255) / 256, 256, 0, stream>>>(inputs, coding, n);
  gemm1_wmma_kernel<<<dim3(BATCH / 16, NPAD / 16), 32, 0, stream>>>(coding, W2, G);
  snn_persistent_kernel<<<BATCH / 16, 512, 0, stream>>>(G, b2, a1_3, a2_3,
                                                        W3, b3, a1_2, a2_2, out);
}